// SPMM_63127429317360
// MI455X (gfx1250) — compile-verified
//
#include <hip/hip_runtime.h>

// ---------------------------------------------------------------------------
// COO SpMM: out[row[e],:] += 0.5 * val[e] * x[col[e],:]   (N=100k, D=64, fp32)
// Bandwidth-bound scatter/gather. CDNA5 path: TDM (tensor_load_to_lds) streams
// the edge lists into LDS double-buffered (TENSORcnt pipeline). Compute layout:
// wave32 split into two half-waves, each covering one 64-float feature row as
// float4 (global_load_b128 gather), scatter via hardware global_atomic_add_f32.
// x (25.6 MB) is L2-resident on MI455X (192 MB L2) so gathers hit L2.
// ---------------------------------------------------------------------------

#define TILE    1024          // edges per TDM tile (per stream: 4 KB)
#define THREADS 256           // 8 waves (wave32)
#define WAVES   8
#define DFEAT   64

typedef unsigned u32x4 __attribute__((ext_vector_type(4)));
typedef int      i32x8 __attribute__((ext_vector_type(8)));
typedef int      i32x4 __attribute__((ext_vector_type(4)));

// LDS byte offset of a __shared__ pointer: generic-AS LDS addresses carry the
// LDS offset in their low 32 bits (CDNA5 aperture mapping).
__device__ __forceinline__ unsigned lds_off_of(const void* p) {
  return (unsigned)(size_t)p;
}

// Issue one TDM 1-D copy: tile_elems x 4-byte elements, global -> LDS.
// tensor_dim0 = elements remaining from gsrc so the partial last tile
// zero-fills (TDM OOB reads return zero).
__device__ __forceinline__ void tdm_load_1d(unsigned lds_byte_off,
                                            const void* gsrc,
                                            unsigned nelem_from_here,
                                            unsigned tile_elems) {
  unsigned long long ga = (unsigned long long)gsrc;
  // D# group 0: count=1 (valid), lds_addr, global_addr[56:0], type=2 ("image")
  u32x4 g0 = { 1u,
               lds_byte_off,
               (unsigned)ga,
               (unsigned)((ga >> 32) & 0x01FFFFFFu) | 0x80000000u };
  // D# group 1: data_size=2 (4B), tensor_dim0 = remaining, tensor_dim1 = 1,
  // tile_dim0 = tile_elems, tile_dim1 = 1, tensor_dim0_stride = remaining.
  i32x8 g1 = { (int)0x00020000u,
               (int)((nelem_from_here & 0xFFFFu) << 16),
               (int)(((nelem_from_here >> 16) & 0xFFFFu) | (1u << 16)),
               (int)((tile_elems & 0xFFFFu) << 16),
               1,
               (int)nelem_from_here,
               0,
               0 };
  i32x4 z4 = { 0, 0, 0, 0 };
#if defined(__clang_major__) && __clang_major__ >= 23
  i32x8 z8 = { 0, 0, 0, 0, 0, 0, 0, 0 };
  __builtin_amdgcn_tensor_load_to_lds(g0, g1, z4, z4, z8, 0);
#else
  __builtin_amdgcn_tensor_load_to_lds(g0, g1, z4, z4, 0);
#endif
}

__device__ __forceinline__ void atomic_add_f32(float* p, float v) {
  __hip_atomic_fetch_add(p, v, __ATOMIC_RELAXED, __HIP_MEMORY_SCOPE_AGENT);
}

// ---------------------------------------------------------------------------
// Kernel 1: zero the output accumulator (harness poisons d_out with 0xAA).
// ---------------------------------------------------------------------------
__global__ void spmm_zero_kernel(float4* __restrict__ out, int n4) {
  int i = blockIdx.x * blockDim.x + threadIdx.x;
  int stride = gridDim.x * blockDim.x;
  float4 z = { 0.f, 0.f, 0.f, 0.f };
  for (; i < n4; i += stride) out[i] = z;
}

// ---------------------------------------------------------------------------
// Kernel 2: TDM-streamed scatter SpMM.
// ---------------------------------------------------------------------------
__global__ void __launch_bounds__(THREADS)
spmm_scatter_kernel(const float* __restrict__ x,
                    const int*   __restrict__ rows,
                    const int*   __restrict__ cols,
                    const float* __restrict__ vals,
                    float*       __restrict__ out,
                    int E, int numTiles) {
  __shared__ int   sRow[2][TILE];
  __shared__ int   sCol[2][TILE];
  __shared__ float sVal[2][TILE];

  const int  tid  = threadIdx.x;
  const int  wv   = tid >> 5;           // wave id 0..7 (wave32)
  const int  ln   = tid & 31;           // lane id
  const int  half = ln >> 4;            // 0: lanes 0-15, 1: lanes 16-31
  const int  sub  = ln & 15;            // lane within half-wave
  const bool lead = (wv == 0);          // wave-uniform: only wave 0 drives TDM
  const int  strideT = gridDim.x;

  int t = blockIdx.x;
  int parity = 0;

  // Prologue: kick off the first tile's 3 streams into buffer 0.
  if (lead && t < numTiles) {
    const unsigned base = (unsigned)t * TILE;
    const unsigned rem  = (unsigned)E - base;
    tdm_load_1d(lds_off_of(&sRow[0][0]), rows + base, rem, TILE);
    tdm_load_1d(lds_off_of(&sCol[0][0]), cols + base, rem, TILE);
    tdm_load_1d(lds_off_of(&sVal[0][0]), vals + base, rem, TILE);
  }

  for (; t < numTiles; t += strideT, parity ^= 1) {
    const int nxt = t + strideT;
    if (lead) {
      if (nxt < numTiles) {
        // Prefetch next tile into the other buffer, then wait for the
        // current tile's 3 loads (TDM ops complete in order per wave):
        // outstanding goes 3 -> 6, wait<=3 drains the first three.
        const unsigned nbase = (unsigned)nxt * TILE;
        const unsigned nrem  = (unsigned)E - nbase;
        const int nb = parity ^ 1;
        tdm_load_1d(lds_off_of(&sRow[nb][0]), rows + nbase, nrem, TILE);
        tdm_load_1d(lds_off_of(&sCol[nb][0]), cols + nbase, nrem, TILE);
        tdm_load_1d(lds_off_of(&sVal[nb][0]), vals + nbase, nrem, TILE);
        __builtin_amdgcn_s_wait_tensorcnt(3);
      } else {
        __builtin_amdgcn_s_wait_tensorcnt(0);
      }
    }
    __syncthreads();   // LDS tile t visible to all waves

    const int base = t * TILE;
    int cnt = E - base;
    if (cnt > TILE) cnt = TILE;

    // Each wave handles two edges per iteration: half-wave (16 lanes x float4)
    // covers one full 64-float feature row with a single b128 gather and four
    // f32 atomic scatters per lane. Loop bound is wave-uniform (p, cnt are
    // uniform); the odd-tail guard (j < cnt) only masks EXEC for one half.
    for (int p = wv; 2 * p < cnt; p += WAVES) {
      const int j = 2 * p + half;
      if (j < cnt) {
        const int   r = sRow[parity][j];
        const int   c = sCol[parity][j];
        const float v = sVal[parity][j] * 0.5f;   // FACTOR fused
        const float4 xv =
            ((const float4*)(x + (size_t)c * DFEAT))[sub];
        float* o = out + (size_t)r * DFEAT + (sub << 2);
        atomic_add_f32(o,     v * xv.x);
        atomic_add_f32(o + 1, v * xv.y);
        atomic_add_f32(o + 2, v * xv.z);
        atomic_add_f32(o + 3, v * xv.w);
      }
    }
    __syncthreads();   // done reading buffer `parity` before it is refilled
  }
}

// ---------------------------------------------------------------------------
// Launch
// ---------------------------------------------------------------------------
extern "C" void kernel_launch(void* const* d_in, const int* in_sizes, int n_in,
                              void* d_out, int out_size, void* d_ws, size_t ws_size,
                              hipStream_t stream) {
  const float* x  = (const float*)d_in[0];   // [N, 64] fp32
  const int*   ei = (const int*)d_in[1];     // [2, E] int32 (row, col)
  const float* ev = (const float*)d_in[2];   // [E] fp32
  float* out = (float*)d_out;                // [N, 64] fp32

  const int E = in_sizes[2];
  const int* rows = ei;                      // TRANSPOSE == False
  const int* cols = ei + E;

  // Zero the accumulator (out_size = N*64, divisible by 4).
  const int n4 = out_size / 4;
  int zgrid = (n4 + 255) / 256;
  if (zgrid > 8192) zgrid = 8192;
  spmm_zero_kernel<<<zgrid, 256, 0, stream>>>((float4*)out, n4);

  const int numTiles = (E + TILE - 1) / TILE;
  int grid = numTiles < 640 ? numTiles : 640;  // persistent blocks; >=2 tiles
  if (grid < 1) grid = 1;                      // each for double-buffering
  spmm_scatter_kernel<<<grid, THREADS, 0, stream>>>(x, rows, cols, ev, out,
                                                    E, numTiles);
}